// Layer_19524921327618
// MI455X (gfx1250) — compile-verified
//
#include <hip/hip_runtime.h>

// ---------------------------------------------------------------------------
// Equivariant GNN layer for MI455X (gfx1250, wave32).
//  Stage 1: zero the 50000x512 f32 accumulator in d_ws (L2-resident: 102MB < 192MB L2)
//  Stage 2: wave-per-edge message build + atomic scatter-add (folds /DENOM)
//  Stage 3: wave-per-16-row-tile GEMMs via V_WMMA_F32_16X16X4_F32 (fp32 WMMA,
//           keeps reference precision; stage is HBM-bandwidth-bound anyway).
//           K-loops fully unrolled & branch-free so the WMMA stream dominates.
// ---------------------------------------------------------------------------

typedef float v2f __attribute__((ext_vector_type(2)));
typedef float v8f __attribute__((ext_vector_type(8)));

#define N_NODES 50000
#define N_EDGES 400000
#define MUL_IN  32
#define MUL_OUT 64
#define AGG_W   512   // 32 + 96 + 160 + 224
#define OUT_W   1024  // 64 + 192 + 320 + 448

// sqrt constants
#define S3   1.7320508075688772f
#define S5   2.2360679774997896f
#define S7   2.6457513110645906f
#define S15  3.8729833462074170f
#define S42  6.4807406984078604f
#define S70  8.3666002653407556f
#define S105 10.246950765959598f

__global__ void zero_agg_kernel(float4* __restrict__ p, int n4) {
    int i = blockIdx.x * blockDim.x + threadIdx.x;
    if (i < n4) p[i] = make_float4(0.f, 0.f, 0.f, 0.f);
}

// One wave (32 lanes) per edge; lane = input channel m.
__global__ void edge_kernel(const float* __restrict__ pos,
                            const float* __restrict__ nfeat,
                            const int*   __restrict__ senders,
                            const int*   __restrict__ receivers,
                            float*       __restrict__ agg) {
    int wave = (blockIdx.x * blockDim.x + threadIdx.x) >> 5;
    int lane = threadIdx.x & 31;
    if (wave >= N_EDGES) return;
    int s = senders[wave];
    int r = receivers[wave];

    float dx = pos[3 * r + 0] - pos[3 * s + 0];
    float dy = pos[3 * r + 1] - pos[3 * s + 1];
    float dz = pos[3 * r + 2] - pos[3 * s + 2];
    float rn  = sqrtf(dx * dx + dy * dy + dz * dz);
    float inv = 1.0f / fmaxf(rn, 1e-12f);
    float x = dx * inv, y = dy * inv, z = dz * inv;
    float x2 = x * x, y2 = y * y, z2 = z * z;

    float sh1[3] = { S3 * x, S3 * y, S3 * z };
    float sh2[5] = { S15 * x * y, S15 * y * z, 0.5f * S5 * (3.0f * z2 - 1.0f),
                     S15 * x * z, 0.5f * S15 * (x2 - y2) };
    float sh3[7] = { 0.25f * S70 * y * (3.0f * x2 - y2),
                     S105 * x * y * z,
                     0.25f * S42 * y * (5.0f * z2 - 1.0f),
                     0.5f  * S7  * z * (5.0f * z2 - 3.0f),
                     0.25f * S42 * x * (5.0f * z2 - 1.0f),
                     0.5f  * S105 * z * (x2 - y2),
                     0.25f * S70 * x * (x2 - 3.0f * y2) };

    // fold the /DENOM (8.0) into the message so the node stage has one uniform scale
    float sf = nfeat[s * MUL_IN + lane] * 0.125f;

    float* b = agg + (size_t)r * AGG_W;
    atomicAdd(b + lane, sf);
#pragma unroll
    for (int i = 0; i < 3; ++i) atomicAdd(b + 32  + lane * 3 + i, sf * sh1[i]);
#pragma unroll
    for (int i = 0; i < 5; ++i) atomicAdd(b + 128 + lane * 5 + i, sf * sh2[i]);
#pragma unroll
    for (int i = 0; i < 7; ++i) atomicAdd(b + 288 + lane * 7 + i, sf * sh3[i]);
}

// One wave per 16-row M-tile. Rows r=(node,i) per irrep; K=32 (plus a second
// K=32 block for the scalar irrep, which absorbs the Wsc skip connection).
// 4 N-tiles of 16 cols -> 4 x v8f accumulators, 32-64 WMMAs per wave.
__global__ void node_wmma_kernel(const float* __restrict__ agg,
                                 const float* __restrict__ nfeat,
                                 const float* __restrict__ W0,
                                 const float* __restrict__ W1,
                                 const float* __restrict__ W2,
                                 const float* __restrict__ W3,
                                 const float* __restrict__ Wsc,
                                 float*       __restrict__ out) {
    const float scale = 0.17677669529663687f;  // 1/sqrt(MUL_IN)

    int wave = (blockIdx.x * blockDim.x + threadIdx.x) >> 5;
    int lane = threadIdx.x & 31;

    // tile ranges: d=1 -> 3125 tiles, d=3 -> 9375, d=5 -> 15625, d=7 -> 21875
    int d, aggbase, outbase, tile0;
    const float* W;
    if (wave < 3125)       { d = 1; aggbase = 0;   outbase = 0;   tile0 = 0;     W = W0; }
    else if (wave < 12500) { d = 3; aggbase = 32;  outbase = 64;  tile0 = 3125;  W = W1; }
    else if (wave < 28125) { d = 5; aggbase = 128; outbase = 256; tile0 = 12500; W = W2; }
    else                   { d = 7; aggbase = 288; outbase = 576; tile0 = 28125; W = W3; }

    int tloc = wave - tile0;
    int mrow = lane & 15;        // A row within tile / B+D column within N-tile
    int hi   = lane >> 4;        // half-wave select
    int koff = hi * 2;           // ISA 16x4 f32 A layout: VGPR0={K0|K2}, VGPR1={K1|K3}

    // A row address: A[r, m] = agg[n*512 + aggbase + m*d + i]
    int r = tloc * 16 + mrow;
    int n = r / d;
    int i = r - n * d;
    const float* arow = agg + (size_t)n * AGG_W + aggbase + i;
    const float* wcol = W + mrow;          // per-lane column base into W

    v8f acc[4] = {};

    // Main K=32 block from the aggregated messages (branch-free, fully unrolled:
    // 8 k-steps x 4 N-tiles = 32 v_wmma_f32_16x16x4_f32)
#pragma unroll
    for (int kk = 0; kk < 8; ++kk) {
        int k = kk * 4 + koff;
        v2f a;
        a.x = arow[(size_t)k * d];
        a.y = arow[(size_t)(k + 1) * d];
#pragma unroll
        for (int c = 0; c < 4; ++c) {
            v2f b;
            b.x = wcol[k * MUL_OUT + c * 16];
            b.y = wcol[(k + 1) * MUL_OUT + c * 16];
            acc[c] = __builtin_amdgcn_wmma_f32_16x16x4_f32(
                false, a, false, b, (short)0, acc[c], false, false);
        }
    }

    // Scalar irrep only: second K=32 block for the skip connection
    // (wave-uniform branch; EXEC stays all-ones inside).
    if (d == 1) {
        const float* nrow = nfeat + (size_t)n * MUL_IN;
        const float* wsc  = Wsc + mrow;
#pragma unroll
        for (int kk = 0; kk < 8; ++kk) {
            int k = kk * 4 + koff;
            v2f a;
            a.x = nrow[k];
            a.y = nrow[k + 1];
#pragma unroll
            for (int c = 0; c < 4; ++c) {
                v2f b;
                b.x = wsc[k * MUL_OUT + c * 16];
                b.y = wsc[(k + 1) * MUL_OUT + c * 16];
                acc[c] = __builtin_amdgcn_wmma_f32_16x16x4_f32(
                    false, a, false, b, (short)0, acc[c], false, false);
            }
        }
    }

    // D layout: VGPR j holds M=j (lanes 0-15) / M=j+8 (lanes 16-31), N=lane%16
    int srow = tloc * 16 + hi * 8;
#pragma unroll
    for (int c = 0; c < 4; ++c) {
#pragma unroll
        for (int j = 0; j < 8; ++j) {
            int rM = srow + j;
            int nn = rM / d;
            int ii = rM - nn * d;
            int t  = c * 16 + mrow;
            out[(size_t)nn * OUT_W + outbase + t * d + ii] = acc[c][j] * scale;
        }
    }
}

extern "C" void kernel_launch(void* const* d_in, const int* in_sizes, int n_in,
                              void* d_out, int out_size, void* d_ws, size_t ws_size,
                              hipStream_t stream) {
    const float* pos  = (const float*)d_in[0];
    const float* nf   = (const float*)d_in[1];
    const float* W0   = (const float*)d_in[2];
    const float* W1   = (const float*)d_in[3];
    const float* W2   = (const float*)d_in[4];
    const float* W3   = (const float*)d_in[5];
    const float* Wsc  = (const float*)d_in[6];
    const int*   send = (const int*)d_in[7];
    const int*   recv = (const int*)d_in[8];
    float* agg = (float*)d_ws;        // 50000*512 f32 = 102.4 MB (L2-resident)
    float* out = (float*)d_out;       // 50000*1024 f32

    int n4 = N_NODES * AGG_W / 4;                       // 6,400,000 float4
    zero_agg_kernel<<<n4 / 256, 256, 0, stream>>>((float4*)agg, n4);

    // one wave per edge, 8 waves per 256-thread block -> exactly 400000 waves
    edge_kernel<<<N_EDGES / 8, 256, 0, stream>>>(pos, nf, send, recv, agg);

    // 50000 M-tiles total across the four irreps -> 6250 blocks of 8 waves
    node_wmma_kernel<<<50000 / 8, 256, 0, stream>>>(agg, nf, W0, W1, W2, W3, Wsc, out);
}